// ClusterNorm2d_38560216383790
// MI455X (gfx1250) — compile-verified
//
#include <hip/hip_runtime.h>

// ---- problem constants (from reference) ----
#define BATCH   64
#define CHANS   64
#define HW      12544          // 112*112
#define HW4     (HW / 4)       // 3136 float4 per (b,c)
#define NBC     (BATCH * CHANS)
#define NCLUST  4
#define EPS_F   1e-5f

// stats kernel decomposition: 12544 = 7 waves * 14 iters * 128 floats
#define SWAVES  7
#define SEG     (HW / SWAVES)  // 1792 floats per wave (contiguous)
#define SITERS  (SEG / 128)    // 14 iterations, compile-time constant

typedef __attribute__((ext_vector_type(2))) float v2f;
typedef __attribute__((ext_vector_type(4))) float v4f;
typedef __attribute__((ext_vector_type(8))) float v8f;

// ---------------------------------------------------------------------------
// Kernel 1: per-(b,c) spatial sum / sum-of-squares via V_WMMA_F32_16X16X4_F32
// with all-ones B (D[m,n] = sum_k A[m,k] + C[m,n]; layout-invariant in B).
// Each wave iteration: one b128 NT load (lane L holds x[4L..4L+3]). The
// (x,y) components across lanes form one complete 16x4 A tile (lane L<16 ->
// row L K0/K1; lane L+16 -> row L K2/K3 per ISA 7.12.2), the (z,w) components
// a second disjoint tile — summation is permutation-invariant, so the two
// tiles cover all 128 floats exactly once. Four independent accumulator
// chains (sum lo/hi, sq lo/hi) for ILP; trip count 14 is uniform across the
// whole block, so EXEC stays all-ones and the loop is a scalar loop.
// D fold: lane L<16 holds rows 0..7 at col L in VGPRs 0..7, lane L+16 rows
// 8..15 -> add 8 regs, then shfl_xor(16), then LDS across 7 waves.
// ---------------------------------------------------------------------------
__global__ __launch_bounds__(224) void cn_stats(const float* __restrict__ x,
                                                float* __restrict__ s_out,
                                                float* __restrict__ ss_out) {
  const int bc   = blockIdx.x;              // 0..4095 = b*CHANS + c
  const int lane = threadIdx.x & 31;
  const int wave = threadIdx.x >> 5;        // 0..6
  const float* xp = x + (size_t)bc * HW + (size_t)wave * SEG + lane * 4;

  v8f acc_s1 = {0.f, 0.f, 0.f, 0.f, 0.f, 0.f, 0.f, 0.f};
  v8f acc_s2 = acc_s1, acc_q1 = acc_s1, acc_q2 = acc_s1;
  const v2f ones = {1.0f, 1.0f};

  #pragma unroll 2
  for (int it = 0; it < SITERS; ++it) {
    v4f d = __builtin_nontemporal_load((const v4f*)(xp + it * 128));
    v2f lo = {d.x, d.y};
    v2f hi = {d.z, d.w};
    v2f lo2 = lo * lo;
    v2f hi2 = hi * hi;
    acc_s1 = __builtin_amdgcn_wmma_f32_16x16x4_f32(false, lo,  false, ones,
                                                   (short)0, acc_s1, false, false);
    acc_s2 = __builtin_amdgcn_wmma_f32_16x16x4_f32(false, hi,  false, ones,
                                                   (short)0, acc_s2, false, false);
    acc_q1 = __builtin_amdgcn_wmma_f32_16x16x4_f32(false, lo2, false, ones,
                                                   (short)0, acc_q1, false, false);
    acc_q2 = __builtin_amdgcn_wmma_f32_16x16x4_f32(false, hi2, false, ones,
                                                   (short)0, acc_q2, false, false);
  }

  v8f as = acc_s1 + acc_s2;
  v8f aq = acc_q1 + acc_q2;
  float ps = as[0] + as[1] + as[2] + as[3] + as[4] + as[5] + as[6] + as[7];
  float pq = aq[0] + aq[1] + aq[2] + aq[3] + aq[4] + aq[5] + aq[6] + aq[7];
  ps += __shfl_xor(ps, 16, 32);             // rows 0..7 + rows 8..15
  pq += __shfl_xor(pq, 16, 32);

  __shared__ float rs[SWAVES], rq[SWAVES];
  if (lane == 0) { rs[wave] = ps; rq[wave] = pq; }
  __syncthreads();
  if (threadIdx.x == 0) {
    float ts = 0.f, tq = 0.f;
    #pragma unroll
    for (int w = 0; w < SWAVES; ++w) { ts += rs[w]; tq += rq[w]; }
    s_out[bc]  = ts;
    ss_out[bc] = tq;
  }
}

// ---------------------------------------------------------------------------
// Kernel 2: cluster segment-sum, unbiased variance, 0.2/0.8 blend, and
// precompute per-(b,c) scale/shift. Single block; negligible runtime.
// ---------------------------------------------------------------------------
__global__ __launch_bounds__(256) void cn_cluster(const float* __restrict__ s,
                                                  const float* __restrict__ ss,
                                                  const float* __restrict__ rmean,
                                                  const float* __restrict__ rvar,
                                                  const float* __restrict__ weight,
                                                  const float* __restrict__ bias,
                                                  const int* __restrict__ labels,
                                                  float* __restrict__ scale,
                                                  float* __restrict__ shift) {
  __shared__ float mg[NCLUST][CHANS];
  __shared__ float vg[NCLUST][CHANS];
  const int tid = threadIdx.x;
  const int g = tid >> 6;          // 0..3
  const int c = tid & 63;          // 0..63

  float S = 0.f, Q = 0.f, cnt = 0.f;
  for (int b = 0; b < BATCH; ++b) {
    if (labels[b] == g) {
      S   += s[b * CHANS + c];
      Q   += ss[b * CHANS + c];
      cnt += 1.f;
    }
  }
  const float N    = cnt * (float)HW;
  const float mean = S / fmaxf(N, 1.f);
  const float var  = (Q - N * mean * mean) / fmaxf(N - 1.f, 1.f);
  mg[g][c] = mean;
  vg[g][c] = var;
  __syncthreads();

  for (int it = 0; it < NBC / 256; ++it) {
    const int idx = it * 256 + tid;        // (b,c) flat
    const int b  = idx >> 6;
    const int cc = idx & 63;
    const int gb = labels[b];
    const float mu  = 0.2f * mg[gb][cc] + 0.8f * rmean[cc];
    const float va  = 0.2f * vg[gb][cc] + 0.8f * rvar[cc];
    const float inv = rsqrtf(va + EPS_F);
    const float sc  = weight[cc] * inv;
    scale[idx] = sc;
    shift[idx] = bias[cc] - mu * sc;
  }
}

// ---------------------------------------------------------------------------
// Kernel 3: out = x*scale[bc] + shift[bc], float4 streams with NT hints.
// blockIdx.y == bc -> scale/shift become scalar (SGPR) loads.
// ---------------------------------------------------------------------------
__global__ __launch_bounds__(256) void cn_norm(const float* __restrict__ x,
                                               const float* __restrict__ scale,
                                               const float* __restrict__ shift,
                                               float* __restrict__ out) {
  const int bc = blockIdx.y;
  const int i  = blockIdx.x * 256 + threadIdx.x;     // float4 index within bc
  if (i >= HW4) return;
  const float sc = scale[bc];
  const float sh = shift[bc];
  const size_t off = (size_t)bc * HW4 + i;
  v4f v = __builtin_nontemporal_load((const v4f*)x + off);
  v4f r;
  r.x = fmaf(v.x, sc, sh);
  r.y = fmaf(v.y, sc, sh);
  r.z = fmaf(v.z, sc, sh);
  r.w = fmaf(v.w, sc, sh);
  __builtin_nontemporal_store(r, (v4f*)out + off);
}

extern "C" void kernel_launch(void* const* d_in, const int* in_sizes, int n_in,
                              void* d_out, int out_size, void* d_ws, size_t ws_size,
                              hipStream_t stream) {
  const float* x      = (const float*)d_in[0];
  const float* rmean  = (const float*)d_in[1];
  const float* rvar   = (const float*)d_in[2];
  const float* weight = (const float*)d_in[3];
  const float* bias   = (const float*)d_in[4];
  const int*   labels = (const int*)d_in[5];
  float* out = (float*)d_out;

  float* ws    = (float*)d_ws;     // 4 * 4096 floats = 64 KB scratch
  float* s     = ws;
  float* ss    = ws + NBC;
  float* scale = ws + 2 * NBC;
  float* shift = ws + 3 * NBC;

  cn_stats<<<NBC, 224, 0, stream>>>(x, s, ss);
  cn_cluster<<<1, 256, 0, stream>>>(s, ss, rmean, rvar, weight, bias, labels,
                                    scale, shift);
  dim3 grid((HW4 + 255) / 256, NBC);   // (13, 4096)
  cn_norm<<<grid, 256, 0, stream>>>(x, scale, shift, out);
}